// DynamicScaledQuantizer_90692529423060
// MI455X (gfx1250) — compile-verified
//
#include <hip/hip_runtime.h>
#include <stdint.h>

// Dynamic per-group (128) asymmetric int8 fake-quantize, fp32 in/out.
// Memory-bound: 512 MB traffic -> ~22us at 23.3 TB/s. One wave32 per group.
// Uses gfx1250 async global->LDS copies (double-buffered) + wave32 shuffle
// reduction. Output stored with non-temporal hint.

typedef float v4f __attribute__((ext_vector_type(4)));

#define GROUP           128
#define WAVES_PER_BLOCK 8
#define BLOCK_THREADS   (WAVES_PER_BLOCK * 32)
#define MAX_BLOCKS      2048

#define QMIN_F (-128.0f)
#define QMAX_F (127.0f)
#define EPS_F  0x1p-23f   // 1.1920928955078125e-07

// Issue an async copy of 16 bytes per lane: global -> LDS.
// ldsByte = per-lane workgroup-relative LDS byte address (low 32 bits of the
// generic pointer to __shared__ memory -- flat LDS aperture keeps the offset
// in addr[31:0]). gptr = per-lane global address (64-bit VGPR pair).
__device__ __forceinline__ void issue_async_group(const float* gptr, uint32_t ldsByte) {
  uint64_t ga = (uint64_t)(uintptr_t)gptr;
  asm volatile("global_load_async_to_lds_b128 %0, %1, off"
               :
               : "v"(ldsByte), "v"(ga)
               : "memory");
}

__global__ __launch_bounds__(BLOCK_THREADS)
void DynamicScaledQuantizer_fq_kernel(const float* __restrict__ x,
                                      float* __restrict__ out,
                                      unsigned totalGroups) {
  __shared__ float smem[2 * WAVES_PER_BLOCK * GROUP];  // 8 KB: 2 buffers/wave

  const unsigned lane   = threadIdx.x & 31u;
  const unsigned wave   = threadIdx.x >> 5;
  const unsigned stride = gridDim.x * WAVES_PER_BLOCK;

  float* slot0 = &smem[(2u * wave + 0u) * GROUP];
  float* slot1 = &smem[(2u * wave + 1u) * GROUP];
  const uint32_t lds0 = (uint32_t)(uintptr_t)(slot0 + 4u * lane);
  const uint32_t lds1 = (uint32_t)(uintptr_t)(slot1 + 4u * lane);

  unsigned g = blockIdx.x * WAVES_PER_BLOCK + wave;
  if (g >= totalGroups) return;

  // Prime the pipeline: copy group g into buffer 0.
  issue_async_group(x + (size_t)g * GROUP + 4u * lane, lds0);

  unsigned cur = 0;
  for (;;) {
    const unsigned gn = g + stride;
    if (gn < totalGroups) {
      // Prefetch the next group into the other buffer, then wait until only
      // that prefetch is outstanding (async loads complete in order).
      issue_async_group(x + (size_t)gn * GROUP + 4u * lane, cur ? lds0 : lds1);
      asm volatile("s_wait_asynccnt 1" ::: "memory");
    } else {
      asm volatile("s_wait_asynccnt 0" ::: "memory");
    }

    const float* s = cur ? slot1 : slot0;
    v4f v = *(const v4f*)(s + 4u * lane);  // ds_load_b128

    // Group min/max: lane-local over 4, then wave32 butterfly reduction.
    float mn = fminf(fminf(v.x, v.y), fminf(v.z, v.w));
    float mx = fmaxf(fmaxf(v.x, v.y), fmaxf(v.z, v.w));
#pragma unroll
    for (int off = 16; off >= 1; off >>= 1) {
      mn = fminf(mn, __shfl_xor(mn, off, 32));
      mx = fmaxf(mx, __shfl_xor(mx, off, 32));
    }
    mn = fminf(mn, 0.0f);  // observer always includes zero
    mx = fmaxf(mx, 0.0f);

    const float scale = fmaxf((mx - mn) / (QMAX_F - QMIN_F), EPS_F);
    float zp = rintf(QMIN_F - mn / scale);          // round-half-even (RNE)
    zp = fminf(fmaxf(zp, QMIN_F), QMAX_F);

    v4f r;
    {
      float q0 = fminf(fmaxf(rintf(v.x / scale) + zp, QMIN_F), QMAX_F);
      float q1 = fminf(fmaxf(rintf(v.y / scale) + zp, QMIN_F), QMAX_F);
      float q2 = fminf(fmaxf(rintf(v.z / scale) + zp, QMIN_F), QMAX_F);
      float q3 = fminf(fmaxf(rintf(v.w / scale) + zp, QMIN_F), QMAX_F);
      r.x = (q0 - zp) * scale;
      r.y = (q1 - zp) * scale;
      r.z = (q2 - zp) * scale;
      r.w = (q3 - zp) * scale;
    }
    // Output is written once and never re-read: non-temporal 128-bit store.
    __builtin_nontemporal_store(r, (v4f*)(out + (size_t)g * GROUP + 4u * lane));

    if (gn >= totalGroups) break;
    g = gn;
    cur ^= 1u;
  }
}

extern "C" void kernel_launch(void* const* d_in, const int* in_sizes, int n_in,
                              void* d_out, int out_size, void* d_ws, size_t ws_size,
                              hipStream_t stream) {
  (void)n_in; (void)out_size; (void)d_ws; (void)ws_size;
  const float* x   = (const float*)d_in[0];
  float*       out = (float*)d_out;

  const unsigned nelem  = (unsigned)in_sizes[0];
  const unsigned groups = nelem / GROUP;   // 8192*8192/128 = 524288
  if (groups == 0) return;

  unsigned needed = (groups + WAVES_PER_BLOCK - 1) / WAVES_PER_BLOCK;
  unsigned blocks = needed < MAX_BLOCKS ? needed : MAX_BLOCKS;

  DynamicScaledQuantizer_fq_kernel<<<blocks, BLOCK_THREADS, 0, stream>>>(x, out, groups);
}